// TemporalAttentionEmbedding_60722247631653
// MI455X (gfx1250) — compile-verified
//
#include <hip/hip_runtime.h>
#include <math.h>

#define NN 50000
#define NE 800000
#define CH 128
#define ED 256

typedef __attribute__((ext_vector_type(2))) float v2f;
typedef __attribute__((ext_vector_type(4))) float v4f;
typedef __attribute__((ext_vector_type(8))) float v8f;
typedef __attribute__((ext_vector_type(4))) int   v4i;

#if __has_builtin(__builtin_amdgcn_global_load_async_to_lds_b128)
#define HAVE_ASYNC_LDS 1
#else
#define HAVE_ASYNC_LDS 0
#endif

// 16-byte global -> LDS copy: async (ASYNCcnt) path when available.
__device__ __forceinline__ void g2lds_b128(const float* g, float* l) {
#if HAVE_ASYNC_LDS
    __builtin_amdgcn_global_load_async_to_lds_b128(
        (__attribute__((address_space(1))) v4i*)g,
        (__attribute__((address_space(3))) v4i*)l, 0, 0);
#else
    *(v4f*)l = *(const v4f*)g;
#endif
}

__device__ __forceinline__ void async_copy_fence() {
#if HAVE_ASYNC_LDS
#if __has_builtin(__builtin_amdgcn_s_wait_asynccnt)
    __builtin_amdgcn_s_wait_asynccnt(0);
#else
    asm volatile("s_wait_asynccnt 0x0" ::: "memory");
#endif
#endif
}

// Order-preserving float->uint encoding so atomicMax(u32) == float max.
__device__ __forceinline__ unsigned fenc(float f) {
    unsigned u = __float_as_uint(f);
    return (u & 0x80000000u) ? ~u : (u | 0x80000000u);
}
__device__ __forceinline__ float fdec(unsigned e) {
    return (e & 0x80000000u) ? __uint_as_float(e & 0x7FFFFFFFu)
                             : __uint_as_float(~e);
}

// ---------------------------------------------------------------------------
// Kernel 1: node-level GEMMs  out = x @ W + b   ([N,128] @ [128,128])
// grid = (N/16, 4): blockIdx.y selects {Q, K, V, skip}; skip writes d_out.
// One wave per 16x16 output tile (8 waves cover 128 cols), f32 WMMA K=4 steps.
// ---------------------------------------------------------------------------
__global__ __launch_bounds__(256) void node_gemm_kernel(
    const float* __restrict__ x,
    const float* __restrict__ Wq, const float* __restrict__ bq,
    const float* __restrict__ Wk, const float* __restrict__ bk,
    const float* __restrict__ Wv, const float* __restrict__ bv,
    const float* __restrict__ Ws, const float* __restrict__ bs,
    float* __restrict__ Qo, float* __restrict__ Ko,
    float* __restrict__ Vo, float* __restrict__ So)
{
    const int mt = blockIdx.x;          // 16-row node tile
    const float* W; const float* bias; float* out;
    switch (blockIdx.y) {
        case 0:  W = Wq; bias = bq; out = Qo; break;
        case 1:  W = Wk; bias = bk; out = Ko; break;
        case 2:  W = Wv; bias = bv; out = Vo; break;
        default: W = Ws; bias = bs; out = So; break;
    }

    __shared__ float xs[16][CH];        // 8 KB tile of x
    const int tid = threadIdx.x;
    #pragma unroll
    for (int q = tid; q < 512; q += 256) {               // 512 x 16B chunks
        const int r = q >> 5, c4 = (q & 31) * 4;
        g2lds_b128(&x[(mt * 16 + r) * CH + c4], &xs[r][c4]);
    }
    async_copy_fence();
    __syncthreads();

    const int wave = tid >> 5, lane = tid & 31;
    const int half = lane >> 4, lm = lane & 15;
    const int n0 = wave * 16;

    v8f c = {0.f, 0.f, 0.f, 0.f, 0.f, 0.f, 0.f, 0.f};
    #pragma unroll
    for (int k0 = 0; k0 < CH; k0 += 4) {
        const int ka = k0 + 2 * half;
        v2f a = *(const v2f*)&xs[lm][ka];                 // A: 16x4 f32 frag
        v2f b;                                            // B: 4x16 f32 frag
        b.x = W[ka * CH + n0 + lm];
        b.y = W[(ka + 1) * CH + n0 + lm];
        c = __builtin_amdgcn_wmma_f32_16x16x4_f32(
                false, a, false, b, (short)0, c, false, false);
    }

    const float bb = bias[n0 + lm];
    #pragma unroll
    for (int i = 0; i < 8; ++i) {
        const int m = mt * 16 + i + 8 * half;             // C/D layout rows
        out[m * CH + n0 + lm] = c[i] + bb;
    }
}

// ---------------------------------------------------------------------------
// Kernel 2: zero segment-max encodings and denominators.
// ---------------------------------------------------------------------------
__global__ __launch_bounds__(256) void init_md_kernel(
    unsigned* __restrict__ m_enc, float* __restrict__ denom, int n2)
{
    const int i = blockIdx.x * 256 + threadIdx.x;
    if (i < n2) { m_enc[i] = 0u; denom[i] = 0.f; }
}

// ---------------------------------------------------------------------------
// Shared helper: build [16 x 256] edge_attr tile (time-encoding || msg) in LDS
// and run the 16x16x4-f32 WMMA chain against We, returning this wave's 16x16
// e_vec tile in `c`.  Wave `w` covers output cols [16w, 16w+16).
// ---------------------------------------------------------------------------
__device__ __forceinline__ v8f edge_evec_tile(
    float (*attr)[ED], const float* __restrict__ rel,
    const float* __restrict__ Wt, const float* __restrict__ bt,
    const float* __restrict__ msg, const float* __restrict__ We,
    int e0, int tid, int n0, int half, int lm)
{
    // msg half: 16 rows x 128 cols = 512 x 16B chunks, async to LDS.
    #pragma unroll
    for (int q = tid; q < 512; q += 256) {
        const int r = q >> 5, c4 = (q & 31) * 4;
        g2lds_b128(&msg[(e0 + r) * CH + c4], &attr[r][CH + c4]);
    }
    // time-encoding half: j is invariant per thread (stride 256 % 128 == 0).
    {
        const int j = tid & 127;
        const float wj = Wt[j], bj = bt[j];
        const int rb = tid >> 7;
        #pragma unroll
        for (int kk = 0; kk < 8; ++kk) {
            const int r = rb + 2 * kk;
            attr[r][j] = cosf(rel[r] * wj + bj);
        }
    }
    async_copy_fence();
    __syncthreads();

    v8f c = {0.f, 0.f, 0.f, 0.f, 0.f, 0.f, 0.f, 0.f};
    #pragma unroll
    for (int k0 = 0; k0 < ED; k0 += 4) {
        const int ka = k0 + 2 * half;
        v2f a = *(const v2f*)&attr[lm][ka];
        v2f b;
        b.x = We[ka * CH + n0 + lm];
        b.y = We[(ka + 1) * CH + n0 + lm];
        c = __builtin_amdgcn_wmma_f32_16x16x4_f32(
                false, a, false, b, (short)0, c, false, false);
    }
    return c;
}

// ---------------------------------------------------------------------------
// Kernel 3: per-edge attention logits + segment max.
// One block per 16 edges; 8 waves cover the 128 e_vec columns.
// ---------------------------------------------------------------------------
__global__ __launch_bounds__(256) void edge_alpha_kernel(
    const float* __restrict__ lu, const float* __restrict__ t,
    const float* __restrict__ msg, const int* __restrict__ eidx,
    const float* __restrict__ Wt, const float* __restrict__ bt,
    const float* __restrict__ We,
    const float* __restrict__ Q, const float* __restrict__ K,
    float* __restrict__ alpha, unsigned* __restrict__ m_enc)
{
    __shared__ float attr[16][ED];
    __shared__ float rel[16];
    __shared__ int   s_src[16], s_dst[16];
    __shared__ float s_alpha[2][16];

    const int tid = threadIdx.x;
    const int e0  = blockIdx.x * 16;

    if (tid < 16) {
        const int e = e0 + tid;
        const int sv = eidx[e], dv = eidx[NE + e];
        s_src[tid] = sv; s_dst[tid] = dv;
        rel[tid] = lu[sv] - t[e];
    }
    if (tid < 32) s_alpha[tid >> 4][tid & 15] = 0.f;
    __syncthreads();

    const int wave = tid >> 5, lane = tid & 31;
    const int half = lane >> 4, lm = lane & 15;
    const int n0 = wave * 16;

    v8f c = edge_evec_tile(attr, rel, Wt, bt, msg, We, e0, tid, n0, half, lm);

    // alpha[e][h] = sum_n q[dst][n] * (k[src][n] + e_vec[n]) over head cols
    const int n = n0 + lm;
    const int h = wave >> 2;                 // cols 0-63 -> head 0, 64-127 -> 1
    #pragma unroll
    for (int i = 0; i < 8; ++i) {
        const int r  = i + 8 * half;
        const int sv = s_src[r], dv = s_dst[r];
        float val = Q[dv * CH + n] * (K[sv * CH + n] + c[i]);
        #pragma unroll
        for (int m = 8; m >= 1; m >>= 1)     // reduce over the 16 col-lanes
            val += __shfl_xor(val, m, 16);
        if (lm == 0) atomicAdd(&s_alpha[h][r], val);
    }
    __syncthreads();

    if (tid < 32) {
        const int h2 = tid >> 4, r = tid & 15;
        const float a = s_alpha[h2][r] * 0.125f;   // 1/sqrt(64)
        alpha[(e0 + r) * 2 + h2] = a;
        atomicMax(&m_enc[s_dst[r] * 2 + h2], fenc(a));
    }
}

// ---------------------------------------------------------------------------
// Kernel 4: p = exp(alpha - m[dst]); denom[dst] += p.  In-place over alpha.
// ---------------------------------------------------------------------------
__global__ __launch_bounds__(256) void edge_p_kernel(
    const int* __restrict__ eidx, const unsigned* __restrict__ m_enc,
    float* __restrict__ alpha_p, float* __restrict__ denom)
{
    const int i = blockIdx.x * 256 + threadIdx.x;
    if (i >= 2 * NE) return;
    const int e = i >> 1, h = i & 1;
    const int dv = eidx[NE + e];
    const float p = expf(alpha_p[i] - fdec(m_enc[dv * 2 + h]));
    alpha_p[i] = p;
    atomicAdd(&denom[dv * 2 + h], p);
}

// ---------------------------------------------------------------------------
// Kernel 5: recompute e_vec, scale (v + e_vec) by attn, scatter-add to out.
// ---------------------------------------------------------------------------
__global__ __launch_bounds__(256) void edge_out_kernel(
    const float* __restrict__ lu, const float* __restrict__ t,
    const float* __restrict__ msg, const int* __restrict__ eidx,
    const float* __restrict__ Wt, const float* __restrict__ bt,
    const float* __restrict__ We,
    const float* __restrict__ V, const float* __restrict__ p,
    const float* __restrict__ denom, float* __restrict__ out)
{
    __shared__ float attr[16][ED];
    __shared__ float rel[16];
    __shared__ int   s_src[16], s_dst[16];
    __shared__ float s_attn[16][2];

    const int tid = threadIdx.x;
    const int e0  = blockIdx.x * 16;

    if (tid < 16) {
        const int e = e0 + tid;
        const int sv = eidx[e], dv = eidx[NE + e];
        s_src[tid] = sv; s_dst[tid] = dv;
        rel[tid] = lu[sv] - t[e];
    }
    __syncthreads();

    if (tid < 32) {
        const int r = tid & 15, h = tid >> 4;
        s_attn[r][h] = p[(e0 + r) * 2 + h] / denom[s_dst[r] * 2 + h];
    }

    const int wave = tid >> 5, lane = tid & 31;
    const int half = lane >> 4, lm = lane & 15;
    const int n0 = wave * 16;

    v8f c = edge_evec_tile(attr, rel, Wt, bt, msg, We, e0, tid, n0, half, lm);

    const int n = n0 + lm;
    const int h = wave >> 2;
    #pragma unroll
    for (int i = 0; i < 8; ++i) {
        const int r  = i + 8 * half;
        const int sv = s_src[r], dv = s_dst[r];
        const float val = s_attn[r][h] * (V[sv * CH + n] + c[i]);
        atomicAdd(&out[dv * CH + n], val);
    }
}

// ---------------------------------------------------------------------------
extern "C" void kernel_launch(void* const* d_in, const int* in_sizes, int n_in,
                              void* d_out, int out_size, void* d_ws, size_t ws_size,
                              hipStream_t stream)
{
    const float* x   = (const float*)d_in[0];
    const float* lu  = (const float*)d_in[1];
    const float* t   = (const float*)d_in[2];
    const float* msg = (const float*)d_in[3];
    const int*   ei  = (const int*)  d_in[4];
    const float* Wt  = (const float*)d_in[5];
    const float* bt  = (const float*)d_in[6];
    const float* Wq  = (const float*)d_in[7];
    const float* bq  = (const float*)d_in[8];
    const float* Wk  = (const float*)d_in[9];
    const float* bk  = (const float*)d_in[10];
    const float* Wv  = (const float*)d_in[11];
    const float* bv  = (const float*)d_in[12];
    const float* We  = (const float*)d_in[13];
    const float* Ws  = (const float*)d_in[14];
    const float* bs  = (const float*)d_in[15];
    float* out = (float*)d_out;

    // Workspace layout (~84 MB): Q | K | V | alpha/p | m_enc | denom
    float*    Qw    = (float*)d_ws;
    float*    Kw    = Qw + (size_t)NN * CH;
    float*    Vw    = Kw + (size_t)NN * CH;
    float*    alpha = Vw + (size_t)NN * CH;
    unsigned* m_enc = (unsigned*)(alpha + (size_t)2 * NE);
    float*    denom = (float*)(m_enc + (size_t)2 * NN);

    // 1) Q/K/V projections into ws, skip connection straight into d_out.
    node_gemm_kernel<<<dim3(NN / 16, 4), 256, 0, stream>>>(
        x, Wq, bq, Wk, bk, Wv, bv, Ws, bs, Qw, Kw, Vw, out);

    // 2) Reset segment max / denom.
    init_md_kernel<<<(2 * NN + 255) / 256, 256, 0, stream>>>(m_enc, denom, 2 * NN);

    // 3) Edge logits + segment max.
    edge_alpha_kernel<<<NE / 16, 256, 0, stream>>>(
        lu, t, msg, ei, Wt, bt, We, Qw, Kw, alpha, m_enc);

    // 4) Softmax numerator + denominator.
    edge_p_kernel<<<(2 * NE + 255) / 256, 256, 0, stream>>>(
        ei, m_enc, alpha, denom);

    // 5) Weighted scatter of (v + e_vec) into output.
    edge_out_kernel<<<NE / 16, 256, 0, stream>>>(
        lu, t, msg, ei, Wt, bt, We, Vw, alpha, denom, out);
}